// DiscreteLayer_2284922602125
// MI455X (gfx1250) — compile-verified
//
#include <hip/hip_runtime.h>

#define BATCH 64
#define DIMS 256
#define SEQ 4096
#define KCODES 32
#define DECAYF 0.99f
#define EPSF 1e-5f

typedef float v2f __attribute__((ext_vector_type(2)));
typedef float v8f __attribute__((ext_vector_type(8)));

__global__ void vq_zero_ws(float* ws, int n) {
  int i = blockIdx.x * blockDim.x + threadIdx.x;
  if (i < n) ws[i] = 0.f;
}

// One wave = one 16-point tile. Two f32 WMMA accumulator chains per wave:
// c0 -> codes 0..15, c1 -> codes 16..31, K-dim stepped 4 at a time (D=256 -> 64 steps).
__global__ __launch_bounds__(256)
void vq_main(const float* __restrict__ x, const float* __restrict__ embed,
             float* __restrict__ out_q, float* __restrict__ g_counts,
             float* __restrict__ g_embed_sum, float* __restrict__ g_loss) {
  __shared__ float unionBuf[DIMS * KCODES];   // phase 1: per-wave score slices; phase 2: block embed_sum
  __shared__ float ek2[KCODES];
  __shared__ float xnormLds[8][32];
  __shared__ int   kminLds[8][16];
  __shared__ float countsLds[KCODES];
  __shared__ float lossLds;

  const int tid  = threadIdx.x;
  const int lane = tid & 31;
  const int w    = tid >> 5;
  const int j    = lane & 15;   // column / point-in-tile
  const int hi   = lane >> 4;   // half-wave select

  // ||e_k||^2 per code, zero block accumulators
  if (tid < KCODES) {
    float s = 0.f;
    for (int d = 0; d < DIMS; ++d) { float e = embed[d * KCODES + tid]; s += e * e; }
    ek2[tid] = s;
    countsLds[tid] = 0.f;
  }
  if (tid == 0) lossLds = 0.f;
  __syncthreads();

  const int b  = blockIdx.x >> 5;                 // 32 blocks per batch row (4096/128)
  const int l0 = ((blockIdx.x & 31) * 8 + w) * 16;
  const float* xb = x + (size_t)b * DIMS * SEQ;

  v8f c0 = {}; v8f c1 = {};
  const int kk = hi * 2;      // A layout 16x4 f32: lanes 0-15 hold K=0,1 ; lanes 16-31 hold K=2,3
  float xnorm = 0.f;
  for (int dbase = 0; dbase < DIMS; dbase += 4) {
    const int r = dbase + kk;
    float a0 = xb[(size_t)(r + 0) * SEQ + l0 + j];
    float a1 = xb[(size_t)(r + 1) * SEQ + l0 + j];
    xnorm += a0 * a0 + a1 * a1;
    v2f av = { a0, a1 };
    // B layout 4x16 f32: VGPR0 = rows K=0 (lanes 0-15) / K=2 (lanes 16-31), VGPR1 = rows K=1 / K=3
    v2f b0 = { embed[(r + 0) * KCODES + j],      embed[(r + 1) * KCODES + j] };
    v2f b1 = { embed[(r + 0) * KCODES + 16 + j], embed[(r + 1) * KCODES + 16 + j] };
    c0 = __builtin_amdgcn_wmma_f32_16x16x4_f32(false, av, false, b0, (short)0, c0, false, false);
    c1 = __builtin_amdgcn_wmma_f32_16x16x4_f32(false, av, false, b1, (short)0, c1, false, false);
  }

  xnormLds[w][lane] = xnorm;
  // D layout 16x16 f32: lane j = col, VGPR v = row (lanes>=16 -> rows 8..15)
  float* score = unionBuf + w * (16 * KCODES);
  const int mrow = hi * 8;
  #pragma unroll
  for (int v = 0; v < 8; ++v) {
    score[(mrow + v) * KCODES + j]      = ek2[j]      - 2.f * c0[v];
    score[(mrow + v) * KCODES + 16 + j] = ek2[16 + j] - 2.f * c1[v];
  }
  __syncthreads();

  if (lane < 16) {
    const float* srow = score + lane * KCODES;
    float best = srow[0]; int bk = 0;
    for (int k = 1; k < KCODES; ++k) { float s = srow[k]; if (s < best) { best = s; bk = k; } }
    kminLds[w][lane] = bk;
    float xn = xnormLds[w][lane] + xnormLds[w][16 + lane];   // full ||x||^2 for point `lane`
    atomicAdd(&countsLds[bk], 1.f);
    atomicAdd(&lossLds, xn + best);                          // min squared distance
  }
  __syncthreads();

  // repurpose unionBuf as block-level embed_sum accumulator
  for (int i = tid; i < DIMS * KCODES; i += 256) unionBuf[i] = 0.f;
  __syncthreads();

  // write quantized output (coalesced per half-wave) + accumulate per-cluster feature sums
  const int km = kminLds[w][j];
  for (int d2 = 0; d2 < DIMS; d2 += 2) {
    const int dr = d2 + hi;
    float xv = xb[(size_t)dr * SEQ + l0 + j];
    float q  = embed[dr * KCODES + km];
    out_q[((size_t)b * DIMS + dr) * SEQ + l0 + j] = q;
    atomicAdd(&unionBuf[dr * KCODES + km], xv);
  }
  __syncthreads();

  for (int i = tid; i < DIMS * KCODES; i += 256) atomicAdd(&g_embed_sum[i], unionBuf[i]);
  if (tid < KCODES) atomicAdd(&g_counts[tid], countsLds[tid]);
  if (tid == 0)     atomicAdd(g_loss, lossLds);
}

__global__ __launch_bounds__(256)
void vq_finalize(const float* __restrict__ cluster_number,
                 const float* __restrict__ embed_avg,
                 const float* __restrict__ embed,
                 const int* __restrict__ training,
                 const float* __restrict__ ws_counts,
                 const float* __restrict__ ws_embed_sum,
                 const float* __restrict__ ws_loss,
                 float* __restrict__ out_embed,
                 float* __restrict__ out_loss) {
  __shared__ float cn[KCODES];
  __shared__ float nsum;
  const int tid = threadIdx.x;
  const int train = training[0];
  if (tid < KCODES) cn[tid] = DECAYF * cluster_number[tid] + (1.f - DECAYF) * ws_counts[tid];
  __syncthreads();
  if (tid == 0) { float s = 0.f; for (int k = 0; k < KCODES; ++k) s += cn[k]; nsum = s; }
  __syncthreads();
  const float n = nsum;
  for (int i = tid; i < DIMS * KCODES; i += 256) {
    if (train) {
      int k = i & (KCODES - 1);
      float ea = DECAYF * embed_avg[i] + (1.f - DECAYF) * ws_embed_sum[i];
      float c  = (cn[k] + EPSF) / (n + KCODES * EPSF) * n;
      out_embed[i] = ea / c;
    } else {
      out_embed[i] = embed[i];
    }
  }
  if (tid == 0) out_loss[0] = ws_loss[0] / (float)((size_t)BATCH * SEQ * DIMS);
}

extern "C" void kernel_launch(void* const* d_in, const int* in_sizes, int n_in,
                              void* d_out, int out_size, void* d_ws, size_t ws_size,
                              hipStream_t stream) {
  (void)in_sizes; (void)n_in; (void)out_size; (void)ws_size;
  const float* x              = (const float*)d_in[0];
  const float* embed          = (const float*)d_in[1];
  const float* cluster_number = (const float*)d_in[2];
  const float* embed_avg      = (const float*)d_in[3];
  const int*   training       = (const int*)d_in[4];

  float* out_q     = (float*)d_out;
  float* out_embed = out_q + (size_t)BATCH * DIMS * SEQ;
  float* out_loss  = out_embed + DIMS * KCODES;

  float* ws           = (float*)d_ws;
  float* ws_counts    = ws;
  float* ws_embed_sum = ws + KCODES;
  float* ws_loss      = ws + KCODES + DIMS * KCODES;
  const int nz = KCODES + DIMS * KCODES + 1;

  vq_zero_ws<<<(nz + 255) / 256, 256, 0, stream>>>(ws, nz);
  vq_main<<<(BATCH * SEQ) / 128, 256, 0, stream>>>(x, embed, out_q, ws_counts,
                                                   ws_embed_sum, ws_loss);
  vq_finalize<<<1, 256, 0, stream>>>(cluster_number, embed_avg, embed, training,
                                     ws_counts, ws_embed_sum, ws_loss,
                                     out_embed, out_loss);
}